// VectorQuantizer_61280593379374
// MI455X (gfx1250) — compile-verified
//
#include <hip/hip_runtime.h>
#include <hip/hip_bf16.h>
#include <math.h>

// ---------------------------------------------------------------------------
// VQ-VAE vector quantizer for MI455X (gfx1250, wave32, WMMA).
//
// dist[n][k] = ||x_n||^2 + ||w_k||^2 - 2 x_n . w_k  via V_WMMA_F32_16X16X4_F32
// (exact fp32 matrix path -> argmin matches the fp32 reference numerics).
// Memory floor is the 268MB one-hot encodings stream; everything is written
// with coalesced wide stores.
// ---------------------------------------------------------------------------

typedef __attribute__((ext_vector_type(2))) float v2f;
typedef __attribute__((ext_vector_type(8))) float v8f;

#define K_CODES   512
#define DIM       64
#define N_VEC     (32 * 64 * 64)        // 131072 vectors
#define Q_ELEMS   (N_VEC * DIM)         // 8388608 quantized elements
#define OUT_Q_OFF 1                     // d_out[0] = loss
#define OUT_PPL_OFF (1 + Q_ELEMS)       // 8388609
#define OUT_ENC_OFF (2 + Q_ELEMS)       // 8388610 (even -> 8B aligned base)
#define WROW      68                    // padded LDS row stride in dwords
                                        // (row*68*4 = row*272 bytes, 16B aligned;
                                        //  bank stride 4 -> conflict-free b64 reads)

// ---------------------------------------------------------------------------
__global__ void vq_init(float* loss_ws, unsigned int* counts) {
    int t = blockIdx.x * blockDim.x + threadIdx.x;
    if (t == 0) *loss_ws = 0.f;
    if (t < K_CODES) counts[t] = 0u;
}

// ---------------------------------------------------------------------------
__global__ __launch_bounds__(256) void vq_main(const float* __restrict__ in,
                                               const float* __restrict__ wgt,
                                               float* __restrict__ out,
                                               float* __restrict__ loss_ws,
                                               unsigned int* __restrict__ counts) {
    __shared__ __align__(16) float wlds[K_CODES * WROW];   // 139264 B codebook
    __shared__ float wnorm[K_CODES];                       //   2048 B ||w_k||^2
    __shared__ float dist_s[8][16][16];                    //   8192 B argmin scratch
    __shared__ int   idx_s[8][16][16];                     //   8192 B
    __shared__ int   kidx_s[8][16];                        //    512 B chosen codes
    __shared__ float xn_s[8][16];                          //    512 B ||x_m||^2

    const int tid = threadIdx.x;

    // ---- stage codebook into padded LDS (512 rows x 16 float4) -------------
    {
        const float4* src = (const float4*)wgt;
        for (int i = tid; i < K_CODES * 16; i += 256) {
            int row = i >> 4, q = i & 15;
            float4 v = src[i];
            *(float4*)&wlds[row * WROW + q * 4] = v;
        }
    }
    __syncthreads();

    // ---- codebook squared norms from LDS -----------------------------------
    for (int c = tid; c < K_CODES; c += 256) {
        float s = 0.f;
        for (int d = 0; d < DIM; ++d) { float w = wlds[c * WROW + d]; s += w * w; }
        wnorm[c] = s;
    }
    __syncthreads();

    const int wave = tid >> 5;
    const int lane = tid & 31;
    const int m    = lane & 15;       // row within 16-row tile
    const int half = lane >> 4;       // 0: K={0,1}, 1: K={2,3} of each k-block
    const int koff = half << 1;

    const int n0 = blockIdx.x * 128 + wave * 16;   // wave owns 16 vectors
    const int n  = n0 + m;
    const int b  = n >> 12;                         // n = b*4096 + h*64 + w
    const int hw = n & 4095;
    const long ibase = (long)b * 262144 + hw;       // NCHW: [b][d][h][w]

    // ---- A fragments: 16 K-blocks of 16x4 fp32, per ISA A-layout -----------
    v2f a[16];
    float xpart = 0.f;
#pragma unroll
    for (int kb = 0; kb < 16; ++kb) {
        int d0 = kb * 4 + koff;
        float x0 = in[ibase + (long)d0 * 4096];
        float x1 = in[ibase + (long)(d0 + 1) * 4096];
        a[kb].x = x0; a[kb].y = x1;
        xpart += x0 * x0 + x1 * x1;
    }
    // combine the two K-halves -> full ||x_m||^2, broadcast via LDS
    float xcomb = xpart + __shfl_xor(xpart, 16);
    if (half == 0) xn_s[wave][m] = xcomb;
    __builtin_amdgcn_wave_barrier();
    float xn[8];
#pragma unroll
    for (int r = 0; r < 8; ++r) xn[r] = xn_s[wave][half * 8 + r];

    float best[8]; int bidx[8];
#pragma unroll
    for (int r = 0; r < 8; ++r) { best[r] = __builtin_inff(); bidx[r] = 0; }

    // ---- 32 code tiles x 16 chained fp32 WMMAs (D=64) ----------------------
    for (int kt = 0; kt < 32; ++kt) {
        const int c = kt * 16 + m;    // this lane's codebook column
        v8f acc = {};
#pragma unroll
        for (int kb = 0; kb < 16; ++kb) {
            int d0 = kb * 4 + koff;
            v2f bf;
            bf.x = wlds[c * WROW + d0];       // B[K][N] = weight[N][K]
            bf.y = wlds[c * WROW + d0 + 1];   // merged -> ds_load_b64, no conflicts
            acc = __builtin_amdgcn_wmma_f32_16x16x4_f32(
                false, a[kb], false, bf, (short)0, acc, false, false);
        }
        const float wn = wnorm[c];
#pragma unroll
        for (int r = 0; r < 8; ++r) {
            float dist = xn[r] + wn - 2.f * acc[r];
            if (dist < best[r]) { best[r] = dist; bidx[r] = c; }
        }
    }

    // ---- cross-lane argmin (16 candidate columns per row) ------------------
#pragma unroll
    for (int r = 0; r < 8; ++r) {
        int rg = half * 8 + r;                 // C layout: lanes16-31 hold rows 8..15
        dist_s[wave][rg][m] = best[r];
        idx_s[wave][rg][m]  = bidx[r];
    }
    __builtin_amdgcn_wave_barrier();
    if (lane < 16) {
        float bd = dist_s[wave][lane][0];
        int   bk = idx_s[wave][lane][0];
        for (int s = 1; s < 16; ++s) {
            float d  = dist_s[wave][lane][s];
            int   k2 = idx_s[wave][lane][s];
            if (d < bd) { bd = d; bk = k2; }
        }
        kidx_s[wave][lane] = bk;
        atomicAdd(&counts[bk], 1u);            // code histogram for perplexity
    }
    __builtin_amdgcn_wave_barrier();

    // ---- quantized straight-through (2x - q) + loss partial ----------------
    // lanes 0-15 write dim d=2*it (16 contiguous floats), lanes 16-31 d=2*it+1
    float lsum = 0.f;
    {
        const int  kk = kidx_s[wave][m];
        const long obase = ibase;              // same (b,hw) as this lane's row
#pragma unroll
        for (int it = 0; it < 32; ++it) {
            int d = it * 2 + half;
            float x = in[obase + (long)d * 4096];
            float q = wlds[kk * WROW + d];
            float diff = q - x;
            lsum += diff * diff;               // loss = (1+0.25)*mean((q-x)^2)
            out[OUT_Q_OFF + obase + (long)d * 4096] = 2.f * x - q;
        }
    }
#pragma unroll
    for (int off = 16; off >= 1; off >>= 1) lsum += __shfl_xor(lsum, off);
    if (lane == 0) atomicAdd(loss_ws, lsum);

    // ---- one-hot encodings: 16 rows x 512 floats, coalesced float2 stream --
    for (int mm = 0; mm < 16; ++mm) {
        const int kk = kidx_s[wave][mm];
        float2* ep = (float2*)(out + (size_t)OUT_ENC_OFF + (size_t)(n0 + mm) * K_CODES);
#pragma unroll
        for (int j = 0; j < 8; ++j) {
            int f2 = j * 32 + lane;
            int c0 = f2 * 2;
            float2 v;
            v.x = (c0     == kk) ? 1.f : 0.f;
            v.y = (c0 + 1 == kk) ? 1.f : 0.f;
            ep[f2] = v;
        }
    }
}

// ---------------------------------------------------------------------------
__global__ __launch_bounds__(512) void vq_final(const float* loss_ws,
                                                const unsigned int* counts,
                                                float* out) {
    __shared__ float red[512];
    int t = threadIdx.x;
    float p = (float)counts[t] / (float)N_VEC;
    red[t] = p * logf(p + 1e-10f);
    __syncthreads();
    for (int s = 256; s > 0; s >>= 1) {
        if (t < s) red[t] += red[t + s];
        __syncthreads();
    }
    if (t == 0) {
        out[OUT_PPL_OFF] = expf(-red[0]);
        out[0] = (*loss_ws) * 1.25f / (float)Q_ELEMS;   // codebook + 0.25*commit
    }
}

// ---------------------------------------------------------------------------
extern "C" void kernel_launch(void* const* d_in, const int* in_sizes, int n_in,
                              void* d_out, int out_size, void* d_ws, size_t ws_size,
                              hipStream_t stream) {
    const float* in  = (const float*)d_in[0];   // [32,64,64,64] fp32 NCHW
    const float* wgt = (const float*)d_in[1];   // [512,64] fp32 codebook
    float* out = (float*)d_out;
    float* loss_ws = (float*)d_ws;
    unsigned int* counts = (unsigned int*)d_ws + 16;

    vq_init<<<1, 512, 0, stream>>>(loss_ws, counts);
    vq_main<<<1024, 256, 0, stream>>>(in, wgt, out, loss_ws, counts);
    vq_final<<<1, 512, 0, stream>>>(loss_ws, counts, out);
}